// Quantizer_25804163514431
// MI455X (gfx1250) — compile-verified
//
#include <hip/hip_runtime.h>
#include <math.h>

// Problem dims (wav2vec2-style quantizer)
#define B_  4
#define S_  2048
#define DM_ 1024          // K
#define G_  8
#define V_  512
#define DG_ 128
#define M_  (B_ * S_)     // 8192 tokens
#define N_  (G_ * V_)     // 4096 logit columns

typedef __attribute__((ext_vector_type(2))) float v2f;
typedef __attribute__((ext_vector_type(8))) float v8f;

#define NEG_INF (-3.402823466e38f)

// ---------------------------------------------------------------------------
// Kernel 0: zero the probability-sum workspace (4096 floats)
// ---------------------------------------------------------------------------
__global__ void quant_zero_ws(float* __restrict__ p, int n) {
  int i = blockIdx.x * blockDim.x + threadIdx.x;
  if (i < n) p[i] = 0.0f;
}

// ---------------------------------------------------------------------------
// Kernel 1: fused GEMM (x@W+b) + gumbel-argmax + codebook lookup + softmax
//           probability accumulation.
// Grid: (M_/32, G_), Block: 256 threads (8 wave32).
// Each workgroup computes a 32x512 logits tile for (token tile, group g):
//   - 8 waves each own a 64-column slice and BOTH 16-row M-halves, so every
//     B fragment (W) load feeds two V_WMMA_F32_16X16X4_F32 ops
//     (10 loads per 8 WMMAs instead of 9 per 4).
//   - Epilogue runs in two 16-row passes over one 32KB LDS tile, with the
//     second pass's accumulators kept live in VGPRs.
// ---------------------------------------------------------------------------
__global__ __launch_bounds__(256) void quant_main(
    const float* __restrict__ x,      // [M_, DM_]
    const float* __restrict__ W,      // [DM_, N_]
    const float* __restrict__ bias,   // [N_]
    const float* __restrict__ CB,     // [G_*V_, DG_]
    const float* __restrict__ U,      // [M_, G_, V_] gumbel uniforms
    float* __restrict__ out,          // [M_, G_*DG_] quantized
    float* __restrict__ probsum)      // [G_, V_] global prob accumulator
{
  __shared__ float s_logits[16][V_];  // 32 KB, reused for both 16-row passes
  __shared__ float s_prob[V_];        // 2 KB

  const int tid  = threadIdx.x;
  const int wave = tid >> 5;
  const int lane = tid & 31;
  const int m0   = blockIdx.x * 32;   // first token of this tile
  const int g    = blockIdx.y;        // group

  // zero the per-WG probability partials
  s_prob[tid]       = 0.0f;
  s_prob[tid + 256] = 0.0f;

  // ------------------------- WMMA GEMM phase ------------------------------
  // A fragment layout (16x4 f32): lanes 0-15 -> M=lane, K=k+{0,1};
  //                               lanes 16-31 -> M=lane-16, K=k+{2,3}.
  // B fragment layout (4x16 f32): lanes 0-15 -> N=col, K=k+{0,1};
  //                               lanes 16-31 -> N=col, K=k+{2,3}.
  // C/D layout (16x16 f32): vgpr r: lanes0-15 M=r, lanes16-31 M=r+8; N=lane&15.
  const int mrow  = lane & 15;
  const int khalf = (lane >> 4) * 2;
  const int nloc  = wave * 64;                     // this wave's 64-col slice

  v8f acc[2][4];                                   // [m-half][n-tile]
  #pragma unroll
  for (int p = 0; p < 2; ++p)
    #pragma unroll
    for (int t = 0; t < 4; ++t)
      #pragma unroll
      for (int r = 0; r < 8; ++r) acc[p][t][r] = 0.0f;

  const float* aRow0   = x + (size_t)(m0 + mrow) * DM_ + khalf;
  const float* aRow1   = aRow0 + (size_t)16 * DM_;
  const int    colbase = g * V_ + nloc + (lane & 15);

  for (int k = 0; k < DM_; k += 4) {
    v2f a0, a1;
    a0.x = aRow0[k];
    a0.y = aRow0[k + 1];
    a1.x = aRow1[k];
    a1.y = aRow1[k + 1];
    #pragma unroll
    for (int t = 0; t < 4; ++t) {
      const float* wp = W + (size_t)(k + khalf) * N_ + colbase + t * 16;
      v2f bb;
      bb.x = wp[0];    // W[k+khalf  ][col]
      bb.y = wp[N_];   // W[k+khalf+1][col]
      acc[0][t] = __builtin_amdgcn_wmma_f32_16x16x4_f32(
          false, a0, false, bb, (short)0, acc[0][t], false, false);
      acc[1][t] = __builtin_amdgcn_wmma_f32_16x16x4_f32(
          false, a1, false, bb, (short)0, acc[1][t], false, false);
    }
  }

  // ----------------- epilogue: two 16-row passes over LDS -----------------
  const int rhalf = (lane >> 4) * 8;
  for (int p = 0; p < 2; ++p) {
    if (p) __syncthreads();   // pass-1 readers done before tile overwrite

    // spill accumulators (+bias) for this 16-row half into LDS
    #pragma unroll
    for (int t = 0; t < 4; ++t) {
      const int cl = nloc + t * 16 + (lane & 15);       // local col 0..511
      const float bv = bias[g * V_ + cl];
      #pragma unroll
      for (int r = 0; r < 8; ++r)
        s_logits[rhalf + r][cl] = acc[p][t][r] + bv;
    }
    __syncthreads();

    // 8 waves x 2 rows each = 16 token rows; 16 cols per lane (v = lane+32i)
    for (int rr = 0; rr < 2; ++rr) {
      const int row   = wave * 2 + rr;                  // 0..15 in LDS tile
      const int token = m0 + p * 16 + row;
      const float* urow = U + ((size_t)token * G_ + g) * V_;

      // pass 1: plain-logit max, noisy-logit argmax
      float lmax = NEG_INF;
      float zmax = NEG_INF;
      int   zarg = 0;
      float lv[16];
      #pragma unroll
      for (int i = 0; i < 16; ++i) {
        const int v = lane + 32 * i;
        const float lg = s_logits[row][v];
        lv[i] = lg;
        lmax = fmaxf(lmax, lg);
        const float u  = urow[v];
        const float gn = -logf(-logf(u));          // gumbel noise
        const float z  = lg + gn;
        if (z > zmax) { zmax = z; zarg = v; }      // first-hit tie (asc. v)
      }
      // wave32 reductions
      #pragma unroll
      for (int off = 16; off >= 1; off >>= 1) {
        lmax = fmaxf(lmax, __shfl_xor(lmax, off, 32));
        const float oz = __shfl_xor(zmax, off, 32);
        const int   oa = __shfl_xor(zarg, off, 32);
        if (oz > zmax || (oz == zmax && oa < zarg)) { zmax = oz; zarg = oa; }
      }

      // pass 2: softmax(plain logits) and LDS prob accumulation
      float ev[16];
      float ssum = 0.0f;
      #pragma unroll
      for (int i = 0; i < 16; ++i) {
        ev[i] = expf(lv[i] - lmax);
        ssum += ev[i];
      }
      #pragma unroll
      for (int off = 16; off >= 1; off >>= 1)
        ssum += __shfl_xor(ssum, off, 32);
      const float inv = 1.0f / ssum;
      #pragma unroll
      for (int i = 0; i < 16; ++i)
        atomicAdd(&s_prob[lane + 32 * i], ev[i] * inv);   // ds_add_f32

      // codebook lookup: quantized[token, g*DG:(g+1)*DG] = CB[g*V + zarg, :]
      const float* cb = CB + ((size_t)g * V_ + zarg) * DG_;
      float* op = out + (size_t)token * (G_ * DG_) + g * DG_;
      #pragma unroll
      for (int d = lane; d < DG_; d += 32)
        op[d] = cb[d];
    }
  }
  __syncthreads();

  // fold per-WG prob partials into global accumulator (one atomic per col)
  for (int v = tid; v < V_; v += 256)
    atomicAdd(&probsum[g * V_ + v], s_prob[v]);
}

// ---------------------------------------------------------------------------
// Kernel 2: perplexity from accumulated probabilities.
// 1 block, 256 threads: wave g handles group g (8 waves == G_).
// ---------------------------------------------------------------------------
__global__ __launch_bounds__(256) void quant_perplexity(
    const float* __restrict__ probsum, float* __restrict__ perp_out)
{
  __shared__ float sred[G_];
  const int wave = threadIdx.x >> 5;
  const int lane = threadIdx.x & 31;
  const float scale = 1.0f / (float)M_;

  float h = 0.0f;
  #pragma unroll
  for (int i = 0; i < 16; ++i) {
    const float p = probsum[wave * V_ + lane + 32 * i] * scale;
    h += p * logf(p + 1e-7f);
  }
  #pragma unroll
  for (int off = 16; off >= 1; off >>= 1)
    h += __shfl_xor(h, off, 32);
  if (lane == 0) sred[wave] = expf(-h);
  __syncthreads();
  if (threadIdx.x == 0) {
    float s = 0.0f;
    #pragma unroll
    for (int g = 0; g < G_; ++g) s += sred[g];
    perp_out[0] = s / (float)G_;
  }
}

// ---------------------------------------------------------------------------
extern "C" void kernel_launch(void* const* d_in, const int* in_sizes, int n_in,
                              void* d_out, int out_size, void* d_ws, size_t ws_size,
                              hipStream_t stream) {
  (void)in_sizes; (void)n_in; (void)out_size; (void)ws_size;
  const float* x    = (const float*)d_in[0];   // [B,S,DM]
  const float* W    = (const float*)d_in[1];   // [DM, G*V]
  const float* bias = (const float*)d_in[2];   // [G*V]
  const float* CB   = (const float*)d_in[3];   // [G*V, DG]
  const float* U    = (const float*)d_in[4];   // [B,S,G,V]
  float* out = (float*)d_out;                  // quantized (8388608) + perp (1)
  float* probsum = (float*)d_ws;               // [G*V] = 16 KB scratch

  quant_zero_ws<<<(N_ + 255) / 256, 256, 0, stream>>>(probsum, N_);

  dim3 grid(M_ / 32, G_);
  quant_main<<<grid, 256, 0, stream>>>(x, W, bias, CB, U, out, probsum);

  quant_perplexity<<<1, 256, 0, stream>>>(probsum, out + (size_t)M_ * G_ * DG_);
}